// Neighbor_Mean_27367531610427
// MI455X (gfx1250) — compile-verified
//
#include <hip/hip_runtime.h>
#include <hip/hip_bf16.h>

typedef __attribute__((ext_vector_type(2))) float v2f;
typedef __attribute__((ext_vector_type(8))) float v8f;

#define B_ 8
#define S_ 2048
#define N_ 32
#define H_ 128
#define LDS_PITCH 132   // 128 + 4 pad -> bank-conflict-free 16x4 A-fragment reads

__global__ __launch_bounds__(256)
void neighbor_mean_wmma_kernel(const float* __restrict__ h,
                               const int*   __restrict__ nidx,
                               const int*   __restrict__ nmask,
                               const float* __restrict__ pos_table,
                               const float* __restrict__ Wn,
                               float*       __restrict__ out)
{
    __shared__ float accLDS[16][LDS_PITCH];

    const int lane = threadIdx.x & 31;
    const int wave = threadIdx.x >> 5;           // 0..7
    const int rowBase = blockIdx.x * 16;         // 16 flattened (b,s) rows per WG

    // ---------------- Phase 1: masked gather-reduce (scaled by 1/N) ----------------
    // Each wave reduces 2 rows; 32 lanes cover H=128 with float4.
    const int c = lane * 4;
    for (int rr = 0; rr < 2; ++rr) {
        const int row = rowBase + wave * 2 + rr;      // flattened b*S + s
        const int b   = row >> 11;                    // row / S_
        const int*   ip = nidx  + (size_t)row * N_;
        const int*   mp = nmask + (size_t)row * N_;
        float4 acc = make_float4(0.f, 0.f, 0.f, 0.f);
        for (int n = 0; n < N_; ++n) {
            const int idx = ip[n];                    // wave-uniform load
            const int msk = mp[n];                    // wave-uniform load
            if (msk) {
                const float4 p = *(const float4*)(pos_table + (size_t)idx * H_ + c);
                acc.x += p.x; acc.y += p.y; acc.z += p.z; acc.w += p.w;
                if (idx > 0) {
                    const float4 hv =
                        *(const float4*)(h + ((size_t)b * S_ + (size_t)(idx - 1)) * H_ + c);
                    acc.x += hv.x; acc.y += hv.y; acc.z += hv.z; acc.w += hv.w;
                }
            }
        }
        const float sc = 1.0f / (float)N_;
        float4* dst = (float4*)&accLDS[wave * 2 + rr][c];   // pitch 528B: 16B aligned
        *dst = make_float4(acc.x * sc, acc.y * sc, acc.z * sc, acc.w * sc);
    }
    __syncthreads();

    // ---------------- Phase 2: out16x128 = acc16x128 @ Wn^T via V_WMMA_F32_16X16X4_F32 ----
    // Wave `wave` produces output columns [wave*16, wave*16+16).
    const int kn   = wave * 16;                  // output col tile base
    const int n16  = lane & 15;                  // M for A-frag, N for B-frag, per ISA layout
    const int hoff = (lane < 16) ? 0 : 2;        // K sub-pair: lanes 0-15 -> {0,1}, 16-31 -> {2,3}

    const float* arow = &accLDS[n16][hoff];                      // A: acc row M=n16
    const float* wrow = Wn + (size_t)(kn + n16) * H_ + hoff;     // B: Wn^T col = Wn row

    v8f cacc = {};
#pragma unroll
    for (int kk = 0; kk < H_ / 4; ++kk) {        // 32 WMMA steps over K=128
        v2f a = *(const v2f*)(arow + kk * 4);    // ds_read_b64, conflict-free via pitch pad
        v2f b = *(const v2f*)(wrow + kk * 4);    // global float2, L2-resident Wn
        cacc = __builtin_amdgcn_wmma_f32_16x16x4_f32(
            /*neg_a=*/false, a, /*neg_b=*/false, b,
            /*c_mod=*/(short)0, cacc, /*reuse_a=*/false, /*reuse_b=*/false);
    }

    // ---------------- Phase 3: store D per 16x16 f32 C/D layout ----------------
    // VGPR j: lanes 0-15 -> M=j, N=lane; lanes 16-31 -> M=j+8, N=lane-16.
#pragma unroll
    for (int j = 0; j < 8; ++j) {
        const int mrow = (lane < 16) ? j : (j + 8);
        out[(size_t)(rowBase + mrow) * H_ + (size_t)(kn + n16)] = cacc[j];
    }
}

extern "C" void kernel_launch(void* const* d_in, const int* in_sizes, int n_in,
                              void* d_out, int out_size, void* d_ws, size_t ws_size,
                              hipStream_t stream) {
    // setup_inputs order: 0:x  1:h  2:g  3:neighbor_index  4:neighbor_mask  5:pos_table  6:Wn
    const float* h_in  = (const float*)d_in[1];
    const int*   nidx  = (const int*)  d_in[3];
    const int*   nmask = (const int*)  d_in[4];
    const float* pos   = (const float*)d_in[5];
    const float* Wn    = (const float*)d_in[6];
    float* out = (float*)d_out;

    const int rows = B_ * S_;                  // 16384
    dim3 grid(rows / 16);                      // 1024 workgroups
    dim3 block(256);                           // 8 waves -> 8 col-tiles of 16
    neighbor_mean_wmma_kernel<<<grid, block, 0, stream>>>(h_in, nidx, nmask, pos, Wn, out);
}